// GAT_12352325943817
// MI455X (gfx1250) — compile-verified
//
#include <hip/hip_runtime.h>
#include <hip/hip_bf16.h>
#include <math.h>

#define N_NODES 50000
#define N_REL   200
#define DIM     64
#define N_HEADS 2
#define OUT_DIM 128
#define N_EDGES 500000
#define N_TRAIN 100000
#define ALPHA   0.2f

typedef __attribute__((ext_vector_type(2))) float v2f;
typedef __attribute__((ext_vector_type(8))) float v8f;

// ---------------------------------------------------------------------------
// Row-wise L2 normalize for 64-wide rows: one wave32 per row, 2 elems/lane.
// ---------------------------------------------------------------------------
__global__ void l2norm64_kernel(const float* __restrict__ in,
                                float* __restrict__ out, int rows) {
  int gid  = blockIdx.x * blockDim.x + threadIdx.x;
  int row  = gid >> 5;           // wave-uniform
  int lane = threadIdx.x & 31;
  if (row >= rows) return;
  float a = in[row * 64 + lane];
  float b = in[row * 64 + 32 + lane];
  float ss = a * a + b * b;
#pragma unroll
  for (int off = 16; off; off >>= 1) ss += __shfl_xor(ss, off, 32);
  float n = fmaxf(sqrtf(ss), 1e-12f);
  out[row * 64 + lane]      = a / n;
  out[row * 64 + 32 + lane] = b / n;
}

// ---------------------------------------------------------------------------
// WMMA f32 GEMM:  C[M x N] = A[M x K] * B
//   WT==1:  B[k][n] = W[n*ldw + woff + k]   (C = A @ W[:, woff:woff+K].T)
//   WT==0:  B[k][n] = W[k*ldw + n]          (row-major K x N slab)
// One wave computes a 16 x 64 strip of C (4 accumulators); the A fragment is
// loaded once per k-step and reused across 4 v_wmma_f32_16x16x4_f32 ops.
// N must be a multiple of 64 (true for all uses: 64 or 128). M guarded.
// Register layouts per CDNA5 ISA 7.12.2:
//   A 16x4 : lane = (k>=2)*16 + m, vgpr = k&1
//   B 4x16 : lane = (k>=2)*16 + n, vgpr = k&1
//   C 16x16: lane = (m>=8)*16 + n, vgpr = m&7
// ---------------------------------------------------------------------------
template <int WT>
__global__ void wmma_gemm_f32_kernel(const float* __restrict__ A,
                                     const float* __restrict__ W,
                                     float* __restrict__ C,
                                     int M, int K,
                                     int lda, int ldw, int woff, int ldc) {
  int tile_m = blockIdx.y * blockDim.y + threadIdx.y;
  if (tile_m * 16 >= M) return;                  // wave-uniform exit
  int lane = threadIdx.x;                        // 0..31
  int half = lane >> 4;                          // 0: K={0,1}, 1: K={2,3}
  int l16  = lane & 15;

  int am = tile_m * 16 + l16;                    // A row this lane loads
  if (am > M - 1) am = M - 1;                    // clamp (stores are guarded)
  int bn0 = blockIdx.x * 64 + l16;               // first B column / W row

  v8f acc0 = {}, acc1 = {}, acc2 = {}, acc3 = {};
  const float* ap = A + (size_t)am * lda + half * 2;
#pragma unroll 2
  for (int k0 = 0; k0 < K; k0 += 4) {
    int ka = k0 + half * 2;
    v2f a = *(const v2f*)(ap + k0);
    v2f b0, b1, b2, b3;
    if (WT) {
      b0 = *(const v2f*)(W + (size_t)(bn0)      * ldw + woff + ka);
      b1 = *(const v2f*)(W + (size_t)(bn0 + 16) * ldw + woff + ka);
      b2 = *(const v2f*)(W + (size_t)(bn0 + 32) * ldw + woff + ka);
      b3 = *(const v2f*)(W + (size_t)(bn0 + 48) * ldw + woff + ka);
    } else {
      const float* w0 = W + (size_t)ka * ldw + bn0;
      const float* w1 = w0 + ldw;
      b0.x = w0[0];  b0.y = w1[0];
      b1.x = w0[16]; b1.y = w1[16];
      b2.x = w0[32]; b2.y = w1[32];
      b3.x = w0[48]; b3.y = w1[48];
    }
    acc0 = __builtin_amdgcn_wmma_f32_16x16x4_f32(false, a, false, b0, (short)0,
                                                 acc0, false, false);
    acc1 = __builtin_amdgcn_wmma_f32_16x16x4_f32(false, a, false, b1, (short)0,
                                                 acc1, false, false);
    acc2 = __builtin_amdgcn_wmma_f32_16x16x4_f32(false, a, false, b2, (short)0,
                                                 acc2, false, false);
    acc3 = __builtin_amdgcn_wmma_f32_16x16x4_f32(false, a, false, b3, (short)0,
                                                 acc3, false, false);
  }

  int cn = blockIdx.x * 64 + l16;
#pragma unroll
  for (int j = 0; j < 8; ++j) {
    int cm = tile_m * 16 + j + half * 8;
    if (cm < M) {
      float* cp = C + (size_t)cm * ldc + cn;
      cp[0]  = acc0[j];
      cp[16] = acc1[j];
      cp[32] = acc2[j];
      cp[48] = acc3[j];
    }
  }
}

// ---------------------------------------------------------------------------
// Layer-1 edge attention, both heads fused: wave per edge, 64 dims = 2/lane.
// e = exp(-leaky_relu(edge_m . a2));  scatter e and e*edge_m by tgt.
// pt/ps are [2][N_NODES][64]; pr is [2][N_REL][64]; rowsum [2][N_NODES];
// hsum [2][N_NODES][64]; a2 is (2, 64).
// ---------------------------------------------------------------------------
__global__ void edge_att1_kernel(const int* __restrict__ tgt,
                                 const int* __restrict__ src,
                                 const int* __restrict__ rtyp,
                                 const float* __restrict__ pt,
                                 const float* __restrict__ ps,
                                 const float* __restrict__ pr,
                                 const float* __restrict__ a2,
                                 float* __restrict__ rowsum,
                                 float* __restrict__ hsum,
                                 int n_edges) {
  int gid  = blockIdx.x * blockDim.x + threadIdx.x;
  int e    = gid >> 5;                    // wave-uniform edge id
  int lane = threadIdx.x & 31;
  if (e >= n_edges) return;
  int t = tgt[e], s = src[e], r = rtyp[e];
#pragma unroll
  for (int h = 0; h < N_HEADS; ++h) {
    const float* pth = pt + (size_t)h * N_NODES * 64 + t * 64;
    const float* psh = ps + (size_t)h * N_NODES * 64 + s * 64;
    const float* prh = pr + (size_t)h * N_REL * 64 + r * 64;
    float m0 = pth[lane]      + psh[lane]      + prh[lane];
    float m1 = pth[32 + lane] + psh[32 + lane] + prh[32 + lane];
    float dot = m0 * a2[h * 64 + lane] + m1 * a2[h * 64 + 32 + lane];
#pragma unroll
    for (int off = 16; off; off >>= 1) dot += __shfl_xor(dot, off, 32);
    float lr = (dot >= 0.0f) ? dot : ALPHA * dot;   // leaky_relu
    float ee = expf(-lr);
    float* hh = hsum + (size_t)h * N_NODES * 64 + t * 64;
    if (lane == 0) atomicAdd(&rowsum[h * N_NODES + t], ee);
    atomicAdd(&hh[lane],      ee * m0);
    atomicAdd(&hh[32 + lane], ee * m1);
  }
}

// ---------------------------------------------------------------------------
// Layer-2 edge attention: 128 dims = 4 per lane.
// ---------------------------------------------------------------------------
__global__ void edge_att2_kernel(const int* __restrict__ tgt,
                                 const int* __restrict__ src,
                                 const int* __restrict__ rtyp,
                                 const float* __restrict__ qt,
                                 const float* __restrict__ qs,
                                 const float* __restrict__ qr,
                                 const float* __restrict__ a2,
                                 float* __restrict__ rowsum,
                                 float* __restrict__ hsum,
                                 int n_edges) {
  int gid  = blockIdx.x * blockDim.x + threadIdx.x;
  int e    = gid >> 5;
  int lane = threadIdx.x & 31;
  if (e >= n_edges) return;
  int t = tgt[e], s = src[e], r = rtyp[e];
  float m[4];
  float dot = 0.0f;
#pragma unroll
  for (int i = 0; i < 4; ++i) {
    int d = lane + i * 32;
    m[i] = qt[t * 128 + d] + qs[s * 128 + d] + qr[r * 128 + d];
    dot += m[i] * a2[d];
  }
#pragma unroll
  for (int off = 16; off; off >>= 1) dot += __shfl_xor(dot, off, 32);
  float lr = (dot >= 0.0f) ? dot : ALPHA * dot;
  float ee = expf(-lr);
  if (lane == 0) atomicAdd(&rowsum[t], ee);
#pragma unroll
  for (int i = 0; i < 4; ++i)
    atomicAdd(&hsum[t * 128 + lane + i * 32], ee * m[i]);
}

// ---------------------------------------------------------------------------
// Concat heads:  x[n][h*64+d] = elu(hsum[h][n][d] / rowsum[h][n])
// ---------------------------------------------------------------------------
__global__ void combine_heads_kernel(const float* __restrict__ hsum,
                                     const float* __restrict__ rowsum,
                                     float* __restrict__ x, int total) {
  int idx = blockIdx.x * blockDim.x + threadIdx.x;
  if (idx >= total) return;
  int n = idx >> 7;          // node
  int c = idx & 127;         // column in OUT_DIM
  int h = c >> 6;            // head
  int d = c & 63;
  float rs = rowsum[h * N_NODES + n];
  rs = (rs == 0.0f) ? 1e-12f : rs;
  float v = hsum[(h * N_NODES + n) * 64 + d] / rs;
  x[idx] = (v > 0.0f) ? v : expm1f(v);   // elu
}

// ---------------------------------------------------------------------------
// mask[train_indices[i,2]] = 1.0  (benign race: all write 1.0)
// ---------------------------------------------------------------------------
__global__ void mask_scatter_kernel(const int* __restrict__ tri,
                                    float* __restrict__ mask, int n_train) {
  int i = blockIdx.x * blockDim.x + threadIdx.x;
  if (i < n_train) mask[tri[i * 3 + 2]] = 1.0f;
}

// ---------------------------------------------------------------------------
// out_entity = l2norm(entW + mask * elu(hsum2/rowsum2)); wave per row, 4/lane.
// ---------------------------------------------------------------------------
__global__ void finalize_entity_kernel(const float* __restrict__ entW,
                                       const float* __restrict__ hsum2,
                                       const float* __restrict__ rowsum2,
                                       const float* __restrict__ mask,
                                       float* __restrict__ out, int rows) {
  int gid  = blockIdx.x * blockDim.x + threadIdx.x;
  int row  = gid >> 5;
  int lane = threadIdx.x & 31;
  if (row >= rows) return;
  float rs = rowsum2[row];
  rs = (rs == 0.0f) ? 1e-12f : rs;
  float mk = mask[row];
  float v[4];
  float ss = 0.0f;
#pragma unroll
  for (int i = 0; i < 4; ++i) {
    int d = lane + i * 32;
    float hv = hsum2[row * 128 + d] / rs;
    hv = (hv > 0.0f) ? hv : expm1f(hv);           // elu
    float val = entW[row * 128 + d] + mk * hv;
    v[i] = val;
    ss += val * val;
  }
#pragma unroll
  for (int off = 16; off; off >>= 1) ss += __shfl_xor(ss, off, 32);
  float n = fmaxf(sqrtf(ss), 1e-12f);
#pragma unroll
  for (int i = 0; i < 4; ++i) out[row * 128 + lane + i * 32] = v[i] / n;
}

// ---------------------------------------------------------------------------
static void launch_gemm(const float* A, const float* W, float* C,
                        int M, int N, int K, int lda, int ldw, int woff,
                        int ldc, int wt, hipStream_t stream) {
  dim3 block(32, 4);
  int tiles_m = (M + 15) / 16;
  dim3 grid(N / 64, (tiles_m + 3) / 4);   // N is 64 or 128 in all uses
  if (wt)
    wmma_gemm_f32_kernel<1><<<grid, block, 0, stream>>>(A, W, C, M, K,
                                                        lda, ldw, woff, ldc);
  else
    wmma_gemm_f32_kernel<0><<<grid, block, 0, stream>>>(A, W, C, M, K,
                                                        lda, ldw, woff, ldc);
}

extern "C" void kernel_launch(void* const* d_in, const int* in_sizes, int n_in,
                              void* d_out, int out_size, void* d_ws, size_t ws_size,
                              hipStream_t stream) {
  const int*   edge_list    = (const int*)d_in[0];
  const int*   tri          = (const int*)d_in[1];
  const float* entity_emb   = (const float*)d_in[2];
  const float* relation_emb = (const float*)d_in[3];
  const float* W_entities   = (const float*)d_in[4];
  const float* att_a        = (const float*)d_in[5];   // (2, 64, 192)
  const float* att_a2       = (const float*)d_in[6];   // (2, 64)
  const float* W_rel        = (const float*)d_in[7];   // (64, 128)
  const float* out_a        = (const float*)d_in[8];   // (128, 384)
  const float* out_a2       = (const float*)d_in[9];   // (128,)

  float* out_entity   = (float*)d_out;                             // 50000 x 128
  float* out_relation = (float*)d_out + (size_t)N_NODES * OUT_DIM; // 200 x 128

  const int* tgt  = edge_list;
  const int* srcv = edge_list + N_EDGES;
  const int* rtyp = edge_list + 2 * N_EDGES;

  // ---- workspace layout (floats), with reuse across phases -----------------
  float* ws = (float*)d_ws;
  size_t o = 0;
  float* ent_n  = ws + o; o += (size_t)N_NODES * DIM;            // kept to end
  float* rel_n  = ws + o; o += 256 * DIM;
  float* pt     = ws + o; o += (size_t)2 * N_NODES * DIM;        // reused as qt
  float* ps     = ws + o; o += (size_t)2 * N_NODES * DIM;        // reused as qs
  float* pr     = ws + o; o += (size_t)2 * 256 * DIM;            // reused as qr
  float* hsum   = ws + o; o += (size_t)2 * N_NODES * DIM;        // reused as hsum2
  float* rowsum = ws + o; o += (size_t)2 * N_NODES;              // reused as rowsum2
  float* x      = ws + o; o += (size_t)N_NODES * OUT_DIM;
  float* entW   = ws + o; o += (size_t)N_NODES * OUT_DIM;
  float* mask   = ws + o; o += (size_t)N_NODES;
  float* qt = pt;     // 50000 x 128  == 2 x 50000 x 64
  float* qs = ps;
  float* qr = pr;     // 200 x 128 fits in 2*256*64
  float* hsum2   = hsum;
  float* rowsum2 = rowsum;

  // ---- phase 0: l2 normalize embeddings ------------------------------------
  l2norm64_kernel<<<(N_NODES * 32 + 255) / 256, 256, 0, stream>>>(
      entity_emb, ent_n, N_NODES);
  l2norm64_kernel<<<(N_REL * 32 + 255) / 256, 256, 0, stream>>>(
      relation_emb, rel_n, N_REL);

  // ---- phase 1: node projections for layer-1 attention (WMMA) --------------
  // att_a[h] is (64, 192); a_t = [:, :64], a_s = [:, 64:128], a_r = [:, 128:192]
  for (int h = 0; h < N_HEADS; ++h) {
    const float* Ah = att_a + (size_t)h * DIM * 3 * DIM;
    launch_gemm(ent_n, Ah, pt + (size_t)h * N_NODES * DIM,
                N_NODES, DIM, DIM, DIM, 3 * DIM, 0,       DIM, 1, stream);
    launch_gemm(ent_n, Ah, ps + (size_t)h * N_NODES * DIM,
                N_NODES, DIM, DIM, DIM, 3 * DIM, DIM,     DIM, 1, stream);
    launch_gemm(rel_n, Ah, pr + (size_t)h * N_REL * DIM,
                N_REL,   DIM, DIM, DIM, 3 * DIM, 2 * DIM, DIM, 1, stream);
  }

  // ---- phase 2: layer-1 edge attention (segment softmax-sum) ---------------
  hipMemsetAsync(rowsum, 0, (size_t)2 * N_NODES * sizeof(float), stream);
  hipMemsetAsync(hsum,   0, (size_t)2 * N_NODES * DIM * sizeof(float), stream);
  edge_att1_kernel<<<(N_EDGES * 32 + 255) / 256, 256, 0, stream>>>(
      tgt, srcv, rtyp, pt, ps, pr, att_a2, rowsum, hsum, N_EDGES);
  combine_heads_kernel<<<((N_NODES * OUT_DIM) + 255) / 256, 256, 0, stream>>>(
      hsum, rowsum, x, N_NODES * OUT_DIM);

  // ---- phase 3: out_relation = rel_n @ W_rel (WMMA) ------------------------
  launch_gemm(rel_n, W_rel, out_relation,
              N_REL, OUT_DIM, DIM, DIM, OUT_DIM, 0, OUT_DIM, 0, stream);

  // ---- phase 4: node projections for layer-2 attention (WMMA) --------------
  // out_a is (128, 384); A_t = [:, :128], A_s = [:, 128:256], A_r = [:, 256:384]
  launch_gemm(x, out_a, qt,
              N_NODES, OUT_DIM, OUT_DIM, OUT_DIM, 3 * OUT_DIM, 0,           OUT_DIM, 1, stream);
  launch_gemm(x, out_a, qs,
              N_NODES, OUT_DIM, OUT_DIM, OUT_DIM, 3 * OUT_DIM, OUT_DIM,     OUT_DIM, 1, stream);
  launch_gemm(out_relation, out_a, qr,
              N_REL,   OUT_DIM, OUT_DIM, OUT_DIM, 3 * OUT_DIM, 2 * OUT_DIM, OUT_DIM, 1, stream);

  // ---- phase 5: layer-2 edge attention -------------------------------------
  hipMemsetAsync(rowsum2, 0, (size_t)N_NODES * sizeof(float), stream);
  hipMemsetAsync(hsum2,   0, (size_t)N_NODES * OUT_DIM * sizeof(float), stream);
  edge_att2_kernel<<<(N_EDGES * 32 + 255) / 256, 256, 0, stream>>>(
      tgt, srcv, rtyp, qt, qs, qr, out_a2, rowsum2, hsum2, N_EDGES);

  // ---- phase 6: entW = ent_n @ W_entities (WMMA); mask; finalize -----------
  launch_gemm(ent_n, W_entities, entW,
              N_NODES, OUT_DIM, DIM, DIM, OUT_DIM, 0, OUT_DIM, 0, stream);
  hipMemsetAsync(mask, 0, (size_t)N_NODES * sizeof(float), stream);
  mask_scatter_kernel<<<(N_TRAIN + 255) / 256, 256, 0, stream>>>(tri, mask, N_TRAIN);
  finalize_entity_kernel<<<(N_NODES * 32 + 255) / 256, 256, 0, stream>>>(
      entW, hsum2, rowsum2, mask, out_entity, N_NODES);
}